// SkeletonPriorEmbedding_515396076435
// MI455X (gfx1250) — compile-verified
//
#include <hip/hip_runtime.h>
#include <hip/hip_bf16.h>

typedef float v2f __attribute__((ext_vector_type(2)));
typedef float v8f __attribute__((ext_vector_type(8)));

#define E 4096
#define J 17
#define NB 16          // bones
#define LN_EPS 1e-5f

// ---------------------------------------------------------------------------
// Block-wide sum reduction (deterministic LDS tree).
// ---------------------------------------------------------------------------
__device__ __forceinline__ float block_reduce_sum(float v, float* sm) {
    const int tid = threadIdx.x;
    sm[tid] = v;
    __syncthreads();
    for (int off = blockDim.x >> 1; off > 0; off >>= 1) {
        if (tid < off) sm[tid] += sm[tid + off];
        __syncthreads();
    }
    float r = sm[0];
    __syncthreads();
    return r;
}

// ---------------------------------------------------------------------------
// Kernel 1: L2-normalize each joint embedding, scale by sqrt(E)=64.
// One block (256 threads) per joint row; 16 elements per thread.
// ---------------------------------------------------------------------------
__global__ __launch_bounds__(256) void joint_norm_kernel(
    const float* __restrict__ in, float* __restrict__ out) {
    __shared__ float sm[256];
    const int row = blockIdx.x;          // 0..16
    const int tid = threadIdx.x;
    float x[16];
    float ss = 0.0f;
#pragma unroll
    for (int i = 0; i < 16; ++i) {
        const int n = i * 256 + tid;
        x[i] = in[(size_t)row * E + n];
        ss += x[i] * x[i];
    }
    ss = block_reduce_sum(ss, sm);
    const float nrm = sqrtf(ss);
    const float sc = 64.0f / fmaxf(nrm, 1e-12f);
#pragma unroll
    for (int i = 0; i < 16; ++i) {
        const int n = i * 256 + tid;
        out[(size_t)row * E + n] = x[i] * sc;
    }
}

// ---------------------------------------------------------------------------
// Kernel 2: gather bone endpoints -> pair_emb [16][2E] (concat along K).
// ---------------------------------------------------------------------------
__global__ __launch_bounds__(256) void pair_gather_kernel(
    const float* __restrict__ jemb, const int* __restrict__ bp,
    float* __restrict__ pair) {
    const int idx = blockIdx.x * 256 + threadIdx.x;   // 16*8192 total
    const int r = idx >> 13;                          // bone
    const int k = idx & 8191;                         // 0..2E-1
    const int j = bp[r * 2 + (k >> 12)];              // endpoint 0 or 1
    pair[idx] = jemb[(size_t)j * E + (k & (E - 1))];
}

// ---------------------------------------------------------------------------
// WMMA fp32 GEMM: P[gridDim.y][16][4096] partial = A[16][K] x W[K][4096]
// One wave owns TWO adjacent 16-wide N strips (32 cols): per K-step the A
// fragment is loaded once and fed to two back-to-back WMMAs (independent
// accumulators -> no D->A/B hazard NOPs). K stepped by 4 with
// v_wmma_f32_16x16x4_f32; K/KCHUNK are compile-time so the chunk loop
// unrolls by 8 with no remainder loop.
//
// Fragment layouts (ISA 7.12.2):
//   A 16x4 : lane -> M=lane&15 ; VGPR{0,1} -> K = 2*(lane>>4) + {0,1}
//   B 4x16 : lane -> N=lane&15 ; VGPR{0,1} -> K = 2*(lane>>4) + {0,1}
//   C 16x16: lane -> N=lane&15 ; VGPR v   -> M = v + 8*(lane>>4)
// ---------------------------------------------------------------------------
#define WPB 4   // waves per block (128 threads)

template <int K, int KCHUNK>
__global__ __launch_bounds__(128) void gemm_wmma_kernel(
    const float* __restrict__ A, const float* __restrict__ W,
    float* __restrict__ P) {
    const int lane = threadIdx.x & 31;
    const int wave = threadIdx.x >> 5;
    const int nl   = lane & 15;
    const int half = lane >> 4;                       // 0 or 1
    const int n0   = (blockIdx.x * WPB + wave) << 5;  // 32 cols per wave
    const int kbase = blockIdx.y * KCHUNK;

    // A: row-major [16][K]; lane's float2 at (m=nl, kbase + k0 + 2*half)
    const float* arow = A + (size_t)nl * K + kbase + 2 * half;
    // B: W row-major [K][E]; lane's pair at rows kbase+k0+2*half, +1
    const float* bptr = W + ((size_t)kbase + 2 * half) * E + n0 + nl;

    v8f acc0 = {};
    v8f acc1 = {};
#pragma unroll 8
    for (int k0 = 0; k0 < KCHUNK; k0 += 4) {
        v2f a = *(const v2f*)(arow + k0);
        v2f b0, b1;
        b0.x = bptr[(size_t)k0 * E];
        b0.y = bptr[(size_t)k0 * E + E];
        b1.x = bptr[(size_t)k0 * E + 16];
        b1.y = bptr[(size_t)k0 * E + E + 16];
        acc0 = __builtin_amdgcn_wmma_f32_16x16x4_f32(
            false, a, false, b0, (short)0, acc0, false, false);
        acc1 = __builtin_amdgcn_wmma_f32_16x16x4_f32(
            false, a, false, b1, (short)0, acc1, false, false);
    }

    float* op = P + (size_t)blockIdx.y * (NB * E) + n0 + nl;
#pragma unroll
    for (int v = 0; v < 8; ++v) {
        op[(size_t)(v + 8 * half) * E]      = acc0[v];
        op[(size_t)(v + 8 * half) * E + 16] = acc1[v];
    }
}

// ---------------------------------------------------------------------------
// Reduce K-split partials + bias, LayerNorm over 4096, optional ReLU.
// One block (256 threads) per row; 16 elements per thread.
// ---------------------------------------------------------------------------
__global__ __launch_bounds__(256) void ln_finish_kernel(
    const float* __restrict__ P, int ksplit,
    const float* __restrict__ bias, const float* __restrict__ g,
    const float* __restrict__ be, float* __restrict__ out, int do_relu) {
    __shared__ float sm[256];
    const int row = blockIdx.x;           // 0..15
    const int tid = threadIdx.x;
    float x[16];
    float s = 0.0f;
#pragma unroll
    for (int i = 0; i < 16; ++i) {
        const int n = i * 256 + tid;
        float v = bias[n];
        for (int p = 0; p < ksplit; ++p)
            v += P[(size_t)p * (NB * E) + (size_t)row * E + n];
        x[i] = v;
        s += v;
    }
    const float mu = block_reduce_sum(s, sm) * (1.0f / (float)E);
    float vs = 0.0f;
#pragma unroll
    for (int i = 0; i < 16; ++i) {
        const float d = x[i] - mu;
        vs += d * d;
    }
    const float var = block_reduce_sum(vs, sm) * (1.0f / (float)E);
    const float rs = rsqrtf(var + LN_EPS);
#pragma unroll
    for (int i = 0; i < 16; ++i) {
        const int n = i * 256 + tid;
        float y = (x[i] - mu) * rs * g[n] + be[n];
        if (do_relu) y = fmaxf(y, 0.0f);
        out[(size_t)row * E + n] = y;
    }
}

// ---------------------------------------------------------------------------
// Launch
// ---------------------------------------------------------------------------
extern "C" void kernel_launch(void* const* d_in, const int* in_sizes, int n_in,
                              void* d_out, int out_size, void* d_ws, size_t ws_size,
                              hipStream_t stream) {
    const float* je  = (const float*)d_in[0];  // [17][4096]
    const float* W1  = (const float*)d_in[1];  // [8192][4096]
    const float* b1  = (const float*)d_in[2];
    const float* g1  = (const float*)d_in[3];
    const float* be1 = (const float*)d_in[4];
    const float* W2  = (const float*)d_in[5];  // [4096][4096]
    const float* b2  = (const float*)d_in[6];
    const float* g2  = (const float*)d_in[7];
    const float* be2 = (const float*)d_in[8];
    const int*   bp  = (const int*)d_in[9];    // [16][2]

    float* out  = (float*)d_out;
    float* jout = out;                 // joint_emb: 17*4096
    float* bf   = out + J * E;         // bone_features: 16*4096

    float* ws   = (float*)d_ws;
    float* pair = ws;                              // 16*8192   = 131072
    float* P1   = pair + NB * 2 * E;               // 4*16*4096 = 262144
    float* h    = P1 + 4 * NB * E;                 // 16*4096   =  65536
    float* P2   = h + NB * E;                      // 4*16*4096 = 262144

    // 1) normalize joints -> d_out (first output)
    joint_norm_kernel<<<J, 256, 0, stream>>>(je, jout);

    // 2) gather bone pair activations [16][8192]
    pair_gather_kernel<<<(NB * 2 * E) / 256, 256, 0, stream>>>(jout, bp, pair);

    // 3) GEMM1: [16x8192]x[8192x4096], K split 4 ways (chunk 2048)
    gemm_wmma_kernel<2 * E, 2048>
        <<<dim3(E / (WPB * 32), 4), WPB * 32, 0, stream>>>(pair, W1, P1);

    // 4) bias + LN + ReLU -> h
    ln_finish_kernel<<<NB, 256, 0, stream>>>(P1, 4, b1, g1, be1, h, 1);

    // 5) GEMM2: [16x4096]x[4096x4096], K split 4 ways (chunk 1024)
    gemm_wmma_kernel<E, 1024>
        <<<dim3(E / (WPB * 32), 4), WPB * 32, 0, stream>>>(h, W2, P2);

    // 6) bias + LN -> bone_features (second output)
    ln_finish_kernel<<<NB, 256, 0, stream>>>(P2, 4, b2, g2, be2, bf, 0);
}